// MessageFactoryConductionHeliumBath_1228360646891
// MI455X (gfx1250) — compile-verified
//
#include <hip/hip_runtime.h>
#include <hip/hip_bf16.h>
#include <math.h>
#include <stdint.h>

// ---------------------------------------------------------------------------
// Edge-scatter heat conduction kernel for MI455X (gfx1250).
// Memory-bound: ~128 MB HBM streaming -> ~5.5us floor @ 23.3 TB/s.
//  - CDNA5 async global->LDS copies (ASYNCcnt) double-buffer the edge streams.
//  - Edge streams are use-once: th:TH_LOAD_NT keeps them from evicting the
//    L2-resident node arrays (T/cp ~800KB) and output vector (~400KB).
//  - Segment sums via hardware global_atomic_add_f32 (unsafeAtomicAdd).
// ---------------------------------------------------------------------------

#define CHUNK    1024                 // edges per block-chunk
#define TPB      256                  // 8 wave32 per block; 4 edges per lane
#define NARR     5                    // L, conductivity, A, src, dst
#define ABYTES   (CHUNK * 4)          // 4096 bytes per array per buffer
#define BUFBYTES (NARR * ABYTES)      // 20480 bytes per buffer

#define WAIT_ASYNC(n) asm volatile("s_wait_asynccnt " #n ::: "memory")

__device__ __forceinline__ void async_copy_b128_nt(unsigned lds_off, const void* gaddr) {
  // GV mode: vdst = LDS byte address, vaddr = 64-bit global address.
  // Non-temporal: streaming, use-once data must not displace hot L2 lines.
  asm volatile("global_load_async_to_lds_b128 %0, %1, off th:TH_LOAD_NT"
               :: "v"(lds_off), "v"((unsigned long long)(uintptr_t)gaddr)
               : "memory");
}

__device__ __forceinline__ float edge_energy(float Ts, float Td, float cs, float cd,
                                             float l, float k, float a, float ts) {
  float dT = Ts - Td;
  dT = dT > 0.0f ? dT : 0.0f;                  // relu(T_src - T_dst)
  float g    = dT / l;                         // gradient (l > 0)
  float Ec   = cbrtf(g * k) * a * ts;          // conduction-limited energy
  float ccp  = (cd * cs) / (cd + cs);          // series thermal capacity
  float Emax = dT * ccp;
  return fminf(Ec, Emax);
}

__global__ void zero_out_kernel(float* __restrict__ out, int n) {
  int i = blockIdx.x * blockDim.x + threadIdx.x;
  if (i < n) out[i] = 0.0f;
}

__global__ __launch_bounds__(TPB)
void edge_chunks_kernel(const float* __restrict__ T,  const float* __restrict__ cp,
                        const float* __restrict__ L,  const float* __restrict__ K,
                        const float* __restrict__ A,  const float* __restrict__ ts_p,
                        const int*   __restrict__ src, const int*  __restrict__ dst,
                        float* __restrict__ out, int nchunks) {
  __shared__ __align__(16) unsigned char smem[2 * BUFBYTES];

  const int tid = threadIdx.x;
  const float ts = ts_p[0];                    // uniform scalar -> s_load

  // Generic LDS pointer: low 32 bits are the LDS byte offset (aperture layout).
  const unsigned lane_lds = (unsigned)(size_t)(&smem[0]) + (unsigned)tid * 16u;

  int c = blockIdx.x;
  if (c >= nchunks) return;

  auto stage = [&](int chunk, int buf) {
    const size_t eoff = (size_t)chunk * CHUNK + (size_t)tid * 4;   // 4 edges/lane
    const unsigned b  = lane_lds + (unsigned)buf * BUFBYTES;
    async_copy_b128_nt(b + 0u * ABYTES, (const void*)(L   + eoff));
    async_copy_b128_nt(b + 1u * ABYTES, (const void*)(K   + eoff));
    async_copy_b128_nt(b + 2u * ABYTES, (const void*)(A   + eoff));
    async_copy_b128_nt(b + 3u * ABYTES, (const void*)(src + eoff));
    async_copy_b128_nt(b + 4u * ABYTES, (const void*)(dst + eoff));
  };

  stage(c, 0);                                  // prologue
  int buf = 0;
  for (; c < nchunks; c += gridDim.x) {
    const int cn = c + (int)gridDim.x;
    if (cn < nchunks) {
      stage(cn, buf ^ 1);                       // prefetch next chunk
      WAIT_ASYNC(5);                            // oldest 5 (current buf) retired
    } else {
      WAIT_ASYNC(0);
    }

    const unsigned char* bb = smem + buf * BUFBYTES + tid * 16;
    const float4 l4 = *(const float4*)(bb + 0 * ABYTES);
    const float4 k4 = *(const float4*)(bb + 1 * ABYTES);
    const float4 a4 = *(const float4*)(bb + 2 * ABYTES);
    const int4   s4 = *(const int4*)  (bb + 3 * ABYTES);
    const int4   d4 = *(const int4*)  (bb + 4 * ABYTES);

    const int*   sp = (const int*)&s4;
    const int*   dp = (const int*)&d4;
    const float* lp = (const float*)&l4;
    const float* kp = (const float*)&k4;
    const float* ap = (const float*)&a4;

#pragma unroll
    for (int j = 0; j < 4; ++j) {
      const int s = sp[j];
      const int d = dp[j];
      // Random 4B gathers -> default RT policy, L2-resident (~800KB total).
      const float E = edge_energy(T[s], T[d], cp[s], cp[d], lp[j], kp[j], ap[j], ts);
      if (E != 0.0f) {                          // relu kills ~half the edges
        unsafeAtomicAdd(&out[d],  E);           // heat_received
        unsafeAtomicAdd(&out[s], -E);           // -heat_sent
      }
    }
    buf ^= 1;
  }
}

// Ragged tail (n_edges % CHUNK); empty for the reference sizes.
__global__ void edge_tail_kernel(const float* __restrict__ T,  const float* __restrict__ cp,
                                 const float* __restrict__ L,  const float* __restrict__ K,
                                 const float* __restrict__ A,  const float* __restrict__ ts_p,
                                 const int*   __restrict__ src, const int*  __restrict__ dst,
                                 float* __restrict__ out, int start, int n_edges) {
  int e = start + blockIdx.x * blockDim.x + threadIdx.x;
  if (e >= n_edges) return;
  const int   s = __builtin_nontemporal_load(src + e);
  const int   d = __builtin_nontemporal_load(dst + e);
  const float l = __builtin_nontemporal_load(L + e);
  const float k = __builtin_nontemporal_load(K + e);
  const float a = __builtin_nontemporal_load(A + e);
  const float E = edge_energy(T[s], T[d], cp[s], cp[d], l, k, a, ts_p[0]);
  if (E != 0.0f) {
    unsafeAtomicAdd(&out[d],  E);
    unsafeAtomicAdd(&out[s], -E);
  }
}

extern "C" void kernel_launch(void* const* d_in, const int* in_sizes, int n_in,
                              void* d_out, int out_size, void* d_ws, size_t ws_size,
                              hipStream_t stream) {
  const float* T    = (const float*)d_in[0];
  const float* cp   = (const float*)d_in[1];
  const float* L    = (const float*)d_in[2];
  const float* K    = (const float*)d_in[3];
  const float* A    = (const float*)d_in[4];
  const float* ts_p = (const float*)d_in[5];
  const int*   src  = (const int*)d_in[6];
  const int*   dst  = (const int*)d_in[7];
  float*       out  = (float*)d_out;

  const int n_nodes = in_sizes[0];
  const int n_edges = in_sizes[2];

  // Output is poisoned by the harness; atomics require a clean zero base.
  zero_out_kernel<<<(n_nodes + 255) / 256, 256, 0, stream>>>(out, n_nodes);

  const int nchunks = n_edges / CHUNK;
  if (nchunks > 0) {
    const int grid = nchunks < 1024 ? nchunks : 1024;  // persistent, >=2 chunks/block
    edge_chunks_kernel<<<grid, TPB, 0, stream>>>(T, cp, L, K, A, ts_p, src, dst,
                                                 out, nchunks);
  }
  const int tail_start = nchunks * CHUNK;
  const int ntail = n_edges - tail_start;
  if (ntail > 0) {
    edge_tail_kernel<<<(ntail + 255) / 256, 256, 0, stream>>>(T, cp, L, K, A, ts_p,
                                                              src, dst, out,
                                                              tail_start, n_edges);
  }
}